// GNNConcatSkipConnections_20547123544330
// MI455X (gfx1250) — compile-verified
//
#include <hip/hip_runtime.h>

typedef __attribute__((ext_vector_type(2))) float v2f;
typedef __attribute__((ext_vector_type(8))) float v8f;

#define EPSV 1e-5f

// ---------------- utility ----------------
__global__ void fill_f32(float* __restrict__ p, float v, long n) {
  long i = (long)blockIdx.x * blockDim.x + threadIdx.x;
  if (i < n) p[i] = v;
}

// ---------------- degree / norm ----------------
__global__ void deg_accum(const int* __restrict__ dst, float* __restrict__ deg, int E1) {
  int e = blockIdx.x * blockDim.x + threadIdx.x;
  if (e < E1) atomicAdd(&deg[dst[e]], 1.0f);
}

__global__ void deg_isqrt(const float* __restrict__ deg, float* __restrict__ dis, int N) {
  int i = blockIdx.x * blockDim.x + threadIdx.x;
  if (i < N) dis[i] = rsqrtf(fmaxf(deg[i], 1.0f));
}

__global__ void edge_norm(const int* __restrict__ src, const int* __restrict__ dst,
                          const float* __restrict__ dis, float* __restrict__ nrm, int E1) {
  int e = blockIdx.x * blockDim.x + threadIdx.x;
  if (e < E1) nrm[e] = dis[src[e]] * dis[dst[e]];
}

// ---------------- batch norm (training-mode batch stats) ----------------
__global__ void bn_stats(const float* __restrict__ x, float* __restrict__ mv, int N, int F) {
  int f = blockIdx.x;            // one block per feature column
  float s = 0.f, s2 = 0.f;
  for (int i = threadIdx.x; i < N; i += blockDim.x) {
    float v = x[(long)i * F + f];
    s += v; s2 += v * v;
  }
  __shared__ float sh[256], sh2[256];
  sh[threadIdx.x] = s; sh2[threadIdx.x] = s2;
  __syncthreads();
  for (int off = blockDim.x >> 1; off > 0; off >>= 1) {
    if ((int)threadIdx.x < off) { sh[threadIdx.x] += sh[threadIdx.x + off]; sh2[threadIdx.x] += sh2[threadIdx.x + off]; }
    __syncthreads();
  }
  if (threadIdx.x == 0) {
    float m = sh[0] / (float)N;
    mv[2 * f]     = m;
    mv[2 * f + 1] = sh2[0] / (float)N - m * m;
  }
}

// Normalize x (row stride F) into y (row stride KP >= F, zero-padded columns).
__global__ void bn_apply_pad(const float* __restrict__ x, float* __restrict__ y,
                             const float* __restrict__ mv, const float* __restrict__ g,
                             const float* __restrict__ b, int F, int KP) {
  int i = blockIdx.x;
  int f = threadIdx.x;
  if (f >= KP) return;
  float r = 0.0f;
  if (f < F) {
    float m = mv[2 * f], v = mv[2 * f + 1];
    r = (x[(long)i * F + f] - m) * rsqrtf(v + EPSV) * g[f] + b[f];
  }
  y[(long)i * KP + f] = r;
}

// Zero-pad + TRANSPOSE: W[Fin x Fout] -> WpT[F16 x KP]  (WpT[f*KP + k] = W[k][f])
__global__ void pad_w_t(const float* __restrict__ W, float* __restrict__ WpT,
                        int Fin, int Fout, int KP, int F16) {
  int idx = blockIdx.x * blockDim.x + threadIdx.x;
  if (idx >= KP * F16) return;
  int f = idx / KP, k = idx % KP;
  WpT[idx] = (k < Fin && f < Fout) ? W[(long)k * Fout + f] : 0.0f;
}

// ---------------- dense GEMM via fp32 WMMA (16x16x4), fully padded ----------------
// G[N x F16] = H[N x KP] @ W[KP x Fout] with W pre-transposed/padded to WpT[F16 x KP].
// One wave per 16-row M tile. A fragments for the whole K are preloaded into
// registers once (KP/4 v2f); each N tile then streams only B fragments (b64 loads).
// No bounds checks in the hot path: EXEC stays all-ones for every WMMA.
template <int KP>
__global__ void gemm_wmma_f32(const float* __restrict__ H, const float* __restrict__ WpT,
                              float* __restrict__ G, int N, int F16) {
  constexpr int KITER = KP / 4;
  int gwave = (int)((blockIdx.x * blockDim.x + threadIdx.x) >> 5);  // wave-uniform
  int lane  = threadIdx.x & 31;
  int mbase = gwave * 16;
  if (mbase >= N) return;                 // wave-uniform guard
  int half = lane >> 4;                   // K-pair select: {0,1} or {2,3}
  int l    = lane & 15;

  const float* hrow = H + (long)(mbase + l) * KP + half * 2;
  v2f a[KITER];
#pragma unroll
  for (int k = 0; k < KITER; ++k) a[k] = *(const v2f*)(hrow + 4 * k);  // b64 loads

  int ntiles = F16 >> 4;
  for (int nt = 0; nt < ntiles; ++nt) {
    const float* wcol = WpT + (long)(nt * 16 + l) * KP + half * 2;
    v8f acc = {};
#pragma unroll
    for (int k = 0; k < KITER; ++k) {
      v2f b = *(const v2f*)(wcol + 4 * k);          // single b64 B-fragment load
      acc = __builtin_amdgcn_wmma_f32_16x16x4_f32(
          false, a[k], false, b, (short)0, acc, false, false);
    }
    float* gp = G + (long)(mbase + half * 8) * F16 + nt * 16 + l;
#pragma unroll
    for (int r = 0; r < 8; ++r) gp[(long)r * F16] = acc[r];   // D: VGPR r -> M=r(+8)
  }
}

// ---------------- edge scatter: out[dst] += w * G[src] ----------------
// One block per edge; threadIdx.x = feature. Edges [0,E1) real; [E1,E1+N) self loops.
__global__ void scatter_edges(const int* __restrict__ src, const int* __restrict__ dst,
                              const float* __restrict__ nrm, const float* __restrict__ dis,
                              const float* __restrict__ G, int gstride,
                              float* __restrict__ out, int E1, int F) {
  int e = blockIdx.x;
  int f = threadIdx.x;
  int s, d; float w;
  if (e < E1) { s = src[e]; d = dst[e]; w = nrm[e]; }
  else        { s = d = e - E1; float di = dis[s]; w = di * di; }
  if (f < F) atomicAdd(&out[(long)d * F + f], w * G[(long)s * gstride + f]);
}

// One block per row; h stride F exact.
__global__ void bias_relu(float* __restrict__ h, const float* __restrict__ b, int F) {
  int i = blockIdx.x;
  int f = threadIdx.x;
  if (f < F) {
    long idx = (long)i * F + f;
    h[idx] = fmaxf(h[idx] + b[f], 0.0f);
  }
}

// ---------------- global mean pool ----------------
__global__ void pool_count(const int* __restrict__ batch, float* __restrict__ cnt, int N) {
  int i = blockIdx.x * blockDim.x + threadIdx.x;
  if (i < N) atomicAdd(&cnt[batch[i]], 1.0f);
}

__global__ void pool_sum(const int* __restrict__ batch, const float* __restrict__ h,
                         float* __restrict__ pooled, int F) {
  int i = blockIdx.x;            // one block per node
  int f = threadIdx.x;
  if (f < F) atomicAdd(&pooled[(long)batch[i] * F + f], h[(long)i * F + f]);
}

__global__ void pool_div(float* __restrict__ pooled, const float* __restrict__ cnt, int F) {
  int g = blockIdx.x;
  int f = threadIdx.x;
  if (f < F) pooled[(long)g * F + f] /= fmaxf(cnt[g], 1.0f);
}

// ---------------- small dense linear (per-thread dot) ----------------
__global__ void linear_small(const float* __restrict__ A, const float* __restrict__ W,
                             const float* __restrict__ b, float* __restrict__ out,
                             int M, int K, int Nout) {
  int idx = blockIdx.x * blockDim.x + threadIdx.x;
  if (idx >= M * Nout) return;
  int m = idx / Nout, n = idx % Nout;
  float acc = b[n];
  for (int k = 0; k < K; ++k) acc += A[(long)m * K + k] * W[(long)k * Nout + n];
  out[idx] = acc;
}

// ---------------- host orchestration ----------------
static inline int cdiv(long a, int b) { return (int)((a + b - 1) / b); }
static inline int r32(int f) { return (f + 31) & ~31; }

extern "C" void kernel_launch(void* const* d_in, const int* in_sizes, int n_in,
                              void* d_out, int out_size, void* d_ws, size_t ws_size,
                              hipStream_t stream) {
  const int F0 = 7, F1 = 71, F2 = 135, F3 = 199, FL1 = 49, FL2 = 2;
  const int KP0 = 8, KP1 = 72, KP2 = 136;          // padded K (mult of 4, even)
  const int G1 = 80, G2 = 144, G3 = 208;           // padded Fout (mult of 16)
  const int N  = in_sizes[0] / F0;                 // 100000
  const int E1 = in_sizes[1] / 2;                  // 1600000
  const int NG = out_size / FL2;                   // 256

  const float* x     = (const float*)d_in[0];
  const int*   src   = (const int*)d_in[1];
  const int*   dst   = (const int*)d_in[1] + E1;
  const int*   batch = (const int*)d_in[2];
  const float* bn0g = (const float*)d_in[3],  *bn0b = (const float*)d_in[4];
  const float* bn1g = (const float*)d_in[5],  *bn1b = (const float*)d_in[6];
  const float* bn2g = (const float*)d_in[7],  *bn2b = (const float*)d_in[8];
  const float* W1 = (const float*)d_in[9],  *b1 = (const float*)d_in[10];
  const float* W2 = (const float*)d_in[11], *b2 = (const float*)d_in[12];
  const float* W3 = (const float*)d_in[13], *b3 = (const float*)d_in[14];
  const float* Wl1 = (const float*)d_in[15], *bl1 = (const float*)d_in[16];
  const float* Wl2 = (const float*)d_in[17], *bl2 = (const float*)d_in[18];
  float* out = (float*)d_out;

  // workspace carve-up (floats)
  float* ws = (float*)d_ws;
  long off = 0;
  float* deg    = ws + off; off += N;
  float* dis    = ws + off; off += N;
  float* nrm    = ws + off; off += E1;
  float* mv     = ws + off; off += 2 * 256;
  float* pooled = ws + off; off += (long)NG * F3;
  float* cnt    = ws + off; off += NG;
  float* lin1   = ws + off; off += (long)NG * FL1;
  float* wpad   = ws + off; off += (long)KP2 * G3; // reused per layer (stream-ordered)
  off = (off + 255) & ~255L;
  float* bufA   = ws + off; off += (long)N * KP2;  // padded GEMM input
  float* bufB   = ws + off; off += (long)N * G3;   // padded GEMM output
  float* bufC   = ws + off; off += (long)N * F3;   // scatter output (exact stride)
  (void)ws_size;

  const int T = 256;
  const int gemmGrid = cdiv((long)(N / 16) * 32, T);  // one wave per 16-row tile

  // ---- degrees (self-loop contributes 1 -> init to 1.0) and edge norms ----
  fill_f32<<<cdiv(N, T), T, 0, stream>>>(deg, 1.0f, N);
  deg_accum<<<cdiv(E1, T), T, 0, stream>>>(dst, deg, E1);
  deg_isqrt<<<cdiv(N, T), T, 0, stream>>>(deg, dis, N);
  edge_norm<<<cdiv(E1, T), T, 0, stream>>>(src, dst, dis, nrm, E1);

  // ---- BN0 on x -> bufA (N x KP0, zero padded) ----
  bn_stats<<<F0, T, 0, stream>>>(x, mv, N, F0);
  bn_apply_pad<<<N, r32(KP0), 0, stream>>>(x, bufA, mv, bn0g, bn0b, F0, KP0);

  // ---- Layer 1 ----
  pad_w_t<<<cdiv(KP0 * G1, T), T, 0, stream>>>(W1, wpad, F0, F1, KP0, G1);
  gemm_wmma_f32<KP0><<<gemmGrid, T, 0, stream>>>(bufA, wpad, bufB, N, G1);
  fill_f32<<<cdiv((long)N * F1, T), T, 0, stream>>>(bufC, 0.0f, (long)N * F1);
  scatter_edges<<<E1 + N, r32(F1), 0, stream>>>(src, dst, nrm, dis, bufB, G1, bufC, E1, F1);
  bias_relu<<<N, r32(F1), 0, stream>>>(bufC, b1, F1);
  bn_stats<<<F1, T, 0, stream>>>(bufC, mv, N, F1);
  bn_apply_pad<<<N, r32(KP1), 0, stream>>>(bufC, bufA, mv, bn1g, bn1b, F1, KP1);

  // ---- Layer 2 ----
  pad_w_t<<<cdiv(KP1 * G2, T), T, 0, stream>>>(W2, wpad, F1, F2, KP1, G2);
  gemm_wmma_f32<KP1><<<gemmGrid, T, 0, stream>>>(bufA, wpad, bufB, N, G2);
  fill_f32<<<cdiv((long)N * F2, T), T, 0, stream>>>(bufC, 0.0f, (long)N * F2);
  scatter_edges<<<E1 + N, r32(F2), 0, stream>>>(src, dst, nrm, dis, bufB, G2, bufC, E1, F2);
  bias_relu<<<N, r32(F2), 0, stream>>>(bufC, b2, F2);
  bn_stats<<<F2, T, 0, stream>>>(bufC, mv, N, F2);
  bn_apply_pad<<<N, r32(KP2), 0, stream>>>(bufC, bufA, mv, bn2g, bn2b, F2, KP2);

  // ---- Layer 3 (no BN after) ----
  pad_w_t<<<cdiv(KP2 * G3, T), T, 0, stream>>>(W3, wpad, F2, F3, KP2, G3);
  gemm_wmma_f32<KP2><<<gemmGrid, T, 0, stream>>>(bufA, wpad, bufB, N, G3);
  fill_f32<<<cdiv((long)N * F3, T), T, 0, stream>>>(bufC, 0.0f, (long)N * F3);
  scatter_edges<<<E1 + N, r32(F3), 0, stream>>>(src, dst, nrm, dis, bufB, G3, bufC, E1, F3);
  bias_relu<<<N, r32(F3), 0, stream>>>(bufC, b3, F3);

  // ---- global mean pool ----
  fill_f32<<<cdiv((long)NG * F3, T), T, 0, stream>>>(pooled, 0.0f, (long)NG * F3);
  fill_f32<<<cdiv(NG, T), T, 0, stream>>>(cnt, 0.0f, NG);
  pool_count<<<cdiv(N, T), T, 0, stream>>>(batch, cnt, N);
  pool_sum<<<N, r32(F3), 0, stream>>>(batch, bufC, pooled, F3);
  pool_div<<<NG, r32(F3), 0, stream>>>(pooled, cnt, F3);

  // ---- head linears ----
  linear_small<<<cdiv((long)NG * FL1, T), T, 0, stream>>>(pooled, Wl1, bl1, lin1, NG, F3, FL1);
  linear_small<<<cdiv((long)NG * FL2, T), T, 0, stream>>>(lin1, Wl2, bl2, out, NG, FL1, FL2);
}